// GAT_9663676416700
// MI455X (gfx1250) — compile-verified
//
#include <hip/hip_runtime.h>
#include <hip/hip_bf16.h>
#include <stdint.h>

typedef __attribute__((ext_vector_type(2))) float v2f;
typedef __attribute__((ext_vector_type(8))) float v8f;

#define N_NODES 50000
#define N_EDGES 800000
#define E_TOT   (N_EDGES + N_NODES)   // 850000 with self-loops
#define NEG_SLOPE 0.2f
#define EPS_F 1e-16f

// ---------------------------------------------------------------- utilities
__global__ void zero_f32(float* __restrict__ p, int n) {
  int i = blockIdx.x * blockDim.x + threadIdx.x;
  if (i < n) p[i] = 0.0f;
}

// Ordered encoding so unsigned atomicMax == float max (works for +/-/inf).
__device__ __forceinline__ unsigned f32_enc(float f) {
  unsigned u = __float_as_uint(f);
  return (u & 0x80000000u) ? ~u : (u | 0x80000000u);
}
__device__ __forceinline__ float f32_dec(unsigned e) {
  unsigned u = (e & 0x80000000u) ? (e ^ 0x80000000u) : ~e;
  return __uint_as_float(u);
}

// ---------------------------------------------------------------- WMMA GEMM
// C[M,N] = A[M,K] @ B[K,N] (+bias, optional relu), all row-major fp32.
// Requires M%16==0, N%16==0, K%4==0. One wave computes one 16x16 C tile.
// grid.x = M/16, grid.y = N/64, block = 128 threads (4 waves).
// A 16x4 layout: lanes 0-15 row M=lane, VGPR0=K0,VGPR1=K1; lanes 16-31 K2/K3.
// B 4x16 layout mirrored: lane%16 = column, VGPR pair = consecutive K rows,
// lane-half selects K offset 0 vs 2.
__global__ void wmma_gemm_f32(const float* __restrict__ A,
                              const float* __restrict__ B,
                              const float* __restrict__ bias,
                              float* __restrict__ C,
                              int M, int N, int K, int relu) {
  int lane = threadIdx.x & 31;
  int wave = threadIdx.x >> 5;
  int tm = blockIdx.x;
  int tn = blockIdx.y * 4 + wave;
  int l15   = lane & 15;
  int khalf = (lane >> 4) << 1;          // 0 or 2
  int rowA  = tm * 16 + l15;
  int col   = tn * 16 + l15;

  const float* __restrict__ ap = A + (size_t)rowA * K;
  v8f acc = {};
  for (int k0 = 0; k0 < K; k0 += 4) {
    int ka = k0 + khalf;
    v2f a, b;
    a.x = ap[ka];
    a.y = ap[ka + 1];
    b.x = B[(size_t)ka * N + col];
    b.y = B[(size_t)(ka + 1) * N + col];
    acc = __builtin_amdgcn_wmma_f32_16x16x4_f32(
        /*neg_a=*/false, a, /*neg_b=*/false, b,
        /*c_mod=*/(short)0, acc, /*reuse_a=*/false, /*reuse_b=*/false);
  }

  float bv = bias ? bias[col] : 0.0f;
  int rbase = tm * 16 + ((lane >> 4) << 3);     // lanes 16-31 hold rows M=8..15
  for (int r = 0; r < 8; ++r) {
    float v = acc[r] + bv;
    if (relu) v = fmaxf(v, 0.0f);
    C[(size_t)(rbase + r) * N + col] = v;
  }
}

// ---------------------------------------------------------------- edge phase
// One thread per (edge, head). Edges [0,N_EDGES) come from edge_index,
// edges [N_EDGES, E_TOT) are self-loops.
__device__ __forceinline__ void edge_ids(int e, const long long* __restrict__ ei,
                                         int& src, int& dst) {
  if (e < N_EDGES) {
    src = (int)ei[e];
    dst = (int)ei[N_EDGES + e];
  } else {
    src = dst = e - N_EDGES;
  }
}

// logit[e,h] = sum_c att[h,c] * leaky_relu(xl[src,h,c] + xr[dst,h,c]); seg-max.
__global__ void edge_logit_max(const float* __restrict__ xl,
                               const float* __restrict__ xr,
                               const float* __restrict__ att,
                               const long long* __restrict__ ei,
                               float* __restrict__ elog,
                               unsigned* __restrict__ mx,
                               int H, int C) {
  long long t = (long long)blockIdx.x * blockDim.x + threadIdx.x;
  if (t >= (long long)E_TOT * H) return;
  int e = (int)(t / H), h = (int)(t - (long long)e * H);
  int src, dst; edge_ids(e, ei, src, dst);
  int HC = H * C;
  const float* __restrict__ pl = xl + (size_t)src * HC + h * C;
  const float* __restrict__ pr = xr + (size_t)dst * HC + h * C;
  const float* __restrict__ pa = att + h * C;
  float s = 0.0f;
  for (int c = 0; c < C; ++c) {
    float v = pl[c] + pr[c];
    v = (v > 0.0f) ? v : NEG_SLOPE * v;
    s += pa[c] * v;
  }
  elog[t] = s;
  atomicMax(&mx[dst * H + h], f32_enc(s));
}

// p = exp(logit - max[dst]); seg-sum. Overwrites elog with p.
__global__ void edge_expsum(float* __restrict__ elog,
                            const unsigned* __restrict__ mx,
                            float* __restrict__ sum,
                            const long long* __restrict__ ei,
                            int H) {
  long long t = (long long)blockIdx.x * blockDim.x + threadIdx.x;
  if (t >= (long long)E_TOT * H) return;
  int e = (int)(t / H), h = (int)(t - (long long)e * H);
  int src, dst; edge_ids(e, ei, src, dst); (void)src;
  float m = f32_dec(mx[dst * H + h]);
  float p = expf(elog[t] - m);
  elog[t] = p;
  atomicAdd(&sum[dst * H + h], p);
}

// accum[dst,h,:] += (p / (sum[dst,h]+eps)) * xl[src,h,:]
__global__ void edge_scatter(const float* __restrict__ elog,
                             const float* __restrict__ sum,
                             const float* __restrict__ xl,
                             const long long* __restrict__ ei,
                             float* __restrict__ accum,
                             int H, int C) {
  long long t = (long long)blockIdx.x * blockDim.x + threadIdx.x;
  if (t >= (long long)E_TOT * H) return;
  int e = (int)(t / H), h = (int)(t - (long long)e * H);
  int src, dst; edge_ids(e, ei, src, dst);
  int HC = H * C;
  float alpha = elog[t] / (sum[dst * H + h] + EPS_F);
  const float* __restrict__ pl = xl + (size_t)src * HC + h * C;
  float* __restrict__ po = accum + (size_t)dst * HC + h * C;
  for (int c = 0; c < C; ++c) atomicAdd(&po[c], alpha * pl[c]);
}

// out = (relu?)(acc + bias), elementwise over [n, D]
__global__ void add_bias_act(const float* __restrict__ acc,
                             const float* __restrict__ bias,
                             float* __restrict__ out,
                             int n, int D, int relu) {
  int i = blockIdx.x * blockDim.x + threadIdx.x;
  if (i >= n * D) return;
  float v = acc[i] + bias[i % D];
  if (relu) v = fmaxf(v, 0.0f);
  out[i] = v;
}

// logits[n,0:2] = h1[n,:] @ Wc2[:,0:2] + bc2  (K=128, N=2: too thin for WMMA)
__global__ void logits_kernel(const float* __restrict__ h1,
                              const float* __restrict__ W,
                              const float* __restrict__ b,
                              float* __restrict__ out) {
  int n = blockIdx.x * blockDim.x + threadIdx.x;
  if (n >= N_NODES) return;
  float s0 = b[0], s1 = b[1];
  const float* __restrict__ p = h1 + (size_t)n * 128;
  for (int k = 0; k < 128; ++k) {
    float v = p[k];
    s0 += v * W[2 * k];
    s1 += v * W[2 * k + 1];
  }
  out[2 * n]     = s0;
  out[2 * n + 1] = s1;
}

// ---------------------------------------------------------------- launch
static inline int cdiv(long long a, int b) { return (int)((a + b - 1) / b); }

extern "C" void kernel_launch(void* const* d_in, const int* in_sizes, int n_in,
                              void* d_out, int out_size, void* d_ws, size_t ws_size,
                              hipStream_t stream) {
  const float*     x    = (const float*)d_in[0];
  const long long* ei   = (const long long*)d_in[1];   // [2, N_EDGES] int64
  const float*     Wl1  = (const float*)d_in[3];       // [128,128]
  const float*     Wr1  = (const float*)d_in[4];
  const float*     att1 = (const float*)d_in[5];       // [8,16]
  const float*     b1   = (const float*)d_in[6];       // [128]
  const float*     Wl2  = (const float*)d_in[7];       // [128,64]
  const float*     Wr2  = (const float*)d_in[8];
  const float*     att2 = (const float*)d_in[9];       // [1,64]
  const float*     b2   = (const float*)d_in[10];      // [64]
  const float*     Wc1  = (const float*)d_in[15];      // [64,128]
  const float*     bc1  = (const float*)d_in[16];      // [128]
  const float*     Wc2  = (const float*)d_in[17];      // [128,2]
  const float*     bc2  = (const float*)d_in[18];      // [2]

  float* out    = (float*)d_out;
  float* emb    = out;                       // [50000, 64]
  float* logits = out + (size_t)N_NODES * 64;// [50000, 2]

  // workspace layout (floats); reuse across phases
  float* ws   = (float*)d_ws;
  float* xl1  = ws + 0;          // 6.40M  (reused: h1c)
  float* xr1  = ws + 6400000;    // 6.40M  (reused: xl2 @+0 / xr2 @+3.2M)
  float* acc1 = ws + 12800000;   // 6.40M  accum1 -> h (in place)
  float* e1   = ws + 19200000;   // 6.80M  elog1 (reused: acc2 @+0 / elog2 @+3.2M)
  unsigned* mx = (unsigned*)(ws + 26000000); // 0.40M
  float* sm   = ws + 26400000;   // 0.40M
  float* h    = acc1;
  float* xl2  = xr1;
  float* xr2  = xr1 + 3200000;
  float* acc2 = e1;
  float* e2   = e1 + 3200000;
  float* h1c  = xl1;

  const int B = 256;
  const long long T1 = (long long)E_TOT * 8;  // (edge, head) layer 1
  const long long T2 = (long long)E_TOT;      // layer 2, H=1

  // ---------------- layer 1: GATv2(128 -> 8x16, concat) + relu -------------
  zero_f32<<<cdiv(6400000, B), B, 0, stream>>>(acc1, 6400000);
  zero_f32<<<cdiv(400000, B),  B, 0, stream>>>((float*)mx, 400000);
  zero_f32<<<cdiv(400000, B),  B, 0, stream>>>(sm, 400000);

  wmma_gemm_f32<<<dim3(N_NODES / 16, 2), 128, 0, stream>>>(x, Wl1, nullptr, xl1,
                                                           N_NODES, 128, 128, 0);
  wmma_gemm_f32<<<dim3(N_NODES / 16, 2), 128, 0, stream>>>(x, Wr1, nullptr, xr1,
                                                           N_NODES, 128, 128, 0);

  edge_logit_max<<<cdiv(T1, B), B, 0, stream>>>(xl1, xr1, att1, ei, e1, mx, 8, 16);
  edge_expsum  <<<cdiv(T1, B), B, 0, stream>>>(e1, mx, sm, ei, 8);
  edge_scatter <<<cdiv(T1, B), B, 0, stream>>>(e1, sm, xl1, ei, acc1, 8, 16);
  add_bias_act <<<cdiv(6400000, B), B, 0, stream>>>(acc1, b1, h, N_NODES, 128, 1);

  // ---------------- layer 2: GATv2(128 -> 64, H=1, mean==identity) ---------
  zero_f32<<<cdiv(3200000, B), B, 0, stream>>>(acc2, 3200000);
  zero_f32<<<cdiv(50000, B),   B, 0, stream>>>((float*)mx, 50000);
  zero_f32<<<cdiv(50000, B),   B, 0, stream>>>(sm, 50000);

  wmma_gemm_f32<<<dim3(N_NODES / 16, 1), 128, 0, stream>>>(h, Wl2, nullptr, xl2,
                                                           N_NODES, 64, 128, 0);
  wmma_gemm_f32<<<dim3(N_NODES / 16, 1), 128, 0, stream>>>(h, Wr2, nullptr, xr2,
                                                           N_NODES, 64, 128, 0);

  edge_logit_max<<<cdiv(T2, B), B, 0, stream>>>(xl2, xr2, att2, ei, e2, mx, 1, 64);
  edge_expsum  <<<cdiv(T2, B), B, 0, stream>>>(e2, mx, sm, ei, 1);
  edge_scatter <<<cdiv(T2, B), B, 0, stream>>>(e2, sm, xl2, ei, acc2, 1, 64);
  add_bias_act <<<cdiv(3200000, B), B, 0, stream>>>(acc2, b2, emb, N_NODES, 64, 0);

  // ---------------- classifier --------------------------------------------
  wmma_gemm_f32<<<dim3(N_NODES / 16, 2), 128, 0, stream>>>(emb, Wc1, bc1, h1c,
                                                           N_NODES, 128, 64, 1);
  logits_kernel<<<cdiv(N_NODES, B), B, 0, stream>>>(h1c, Wc2, bc2, logits);
}